// DensityAwareChamferDistance_27307402068673
// MI455X (gfx1250) — compile-verified
//
#include <hip/hip_runtime.h>
#include <math.h>

// ---------------------------------------------------------------------------
// Density-aware Chamfer distance, B=2, D=3, N=M=8192.
// Strategy: P = xx + yy - 2*zz ; zz tile via V_WMMA_F32_16X16X4_F32 (K padded
// 3->4 with zeros). One wave owns 16 x-rows, streams gt columns in 16-wide
// tiles, keeps per-lane running (min,argmin), reduces across 16 lanes with
// wave32 shuffles. Same kernel runs both directions (args swapped).
// Loss math done in double to mirror the reference's float64 upcast.
// ---------------------------------------------------------------------------

typedef float v2f __attribute__((ext_vector_type(2)));
typedef float v8f __attribute__((ext_vector_type(8)));

#define DACD_ALPHA 10.0
#define DACD_EPS   1e-06

// X: [3,N] plane of one batch laid out [B,3,N]; find for each x-point its
// nearest neighbor among Y's M points. dist/idx: [B,N].
__global__ void dacd_nn_wmma(const float* __restrict__ X,
                             const float* __restrict__ Y,
                             float* __restrict__ dist,
                             int* __restrict__ idx,
                             int B, int N, int M) {
  const int wavesPerBlock = blockDim.x >> 5;
  const int wave = blockIdx.x * wavesPerBlock + (threadIdx.x >> 5);
  const int tilesPerBatch = N >> 4;                 // 16 rows per wave
  if (wave >= B * tilesPerBatch) return;            // whole-wave exit: EXEC all-1 for survivors

  const int b       = wave / tilesPerBatch;
  const int rowTile = wave - b * tilesPerBatch;
  const int lane    = threadIdx.x & 31;
  const int half    = lane >> 4;                    // 0: K=0,1 / rows 0-7 ; 1: K=2,3 / rows 8-15
  const int l16     = lane & 15;

  const float* __restrict__ Xb = X + (size_t)b * 3 * N;
  const float* __restrict__ Yb = Y + (size_t)b * 3 * M;

  // A-matrix operand (16x4 f32): lane holds row = rowTile*16 + l16,
  // low half supplies K=0,1 ; high half supplies K=2 and zero-padded K=3.
  const int arow = rowTile * 16 + l16;
  v2f a;
  if (half == 0) { a.x = Xb[arow];         a.y = Xb[N + arow]; }
  else           { a.x = Xb[2 * N + arow]; a.y = 0.0f;         }

  // Per-lane running min over (yy - 2*zz); xx folded in after (argmin invariant).
  float best[8];
  int   bidx[8];
#pragma unroll
  for (int v = 0; v < 8; ++v) { best[v] = 3.402823466e+38f; bidx[v] = 0; }

  const int nColTiles = M >> 4;
  for (int ct = 0; ct < nColTiles; ++ct) {
    const int col = ct * 16 + l16;                  // this lane's column slot
    const float y0 = Yb[col];
    const float y1 = Yb[M + col];
    const float y2 = Yb[2 * M + col];
    if (ct + 1 < nColTiles) {                       // uniform branch; emits global_prefetch_b8
      __builtin_prefetch(&Yb[col + 16], 0, 1);
      __builtin_prefetch(&Yb[M + col + 16], 0, 1);
      __builtin_prefetch(&Yb[2 * M + col + 16], 0, 1);
    }
    // B-matrix operand (4x16 f32): low half K=0,1 ; high half K=2, pad K=3.
    v2f bm;
    bm.x = half ? y2   : y0;
    bm.y = half ? 0.0f : y1;
    const float yy = y0 * y0 + y1 * y1 + y2 * y2;

    v8f c = {};
    // D = A x B + 0 : c[v] = zz[rowTile*16 + v + 8*half][ct*16 + l16]
    c = __builtin_amdgcn_wmma_f32_16x16x4_f32(
        /*neg_a=*/false, a, /*neg_b=*/false, bm,
        /*c_mod=*/(short)0, c, /*reuse_a=*/false, /*reuse_b=*/false);

#pragma unroll
    for (int v = 0; v < 8; ++v) {
      const float cand = yy - 2.0f * c[v];
      if (cand < best[v]) { best[v] = cand; bidx[v] = col; }
    }
  }

  // Min+argmin across the 16 lanes of each half (xor masks keep halves apart).
#pragma unroll
  for (int off = 8; off >= 1; off >>= 1) {
#pragma unroll
    for (int v = 0; v < 8; ++v) {
      const float od = __shfl_xor(best[v], off, 32);
      const int   oi = __shfl_xor(bidx[v], off, 32);
      if (od < best[v] || (od == best[v] && oi < bidx[v])) {
        best[v] = od; bidx[v] = oi;
      }
    }
  }

  if (l16 == 0) {                                   // lanes 0 and 16 own rows 0-7 / 8-15
#pragma unroll
    for (int v = 0; v < 8; ++v) {
      const int r = rowTile * 16 + v + 8 * half;
      const float x0 = Xb[r], x1 = Xb[N + r], x2 = Xb[2 * N + r];
      const float xx = x0 * x0 + x1 * x1 + x2 * x2;
      dist[(size_t)b * N + r] = best[v] + xx;
      idx [(size_t)b * N + r] = bidx[v];
    }
  }
}

__global__ void dacd_init(int* __restrict__ c1, int* __restrict__ c2,
                          double* __restrict__ sums, int n1, int n2) {
  const int i = blockIdx.x * blockDim.x + threadIdx.x;
  if (i < n1) c1[i] = 0;
  if (i < n2) c2[i] = 0;
  if (i < 2)  sums[i] = 0.0;
}

// count[b, idx[b,n]] += 1  over all n (Npts source points, Mtgt targets)
__global__ void dacd_count(const int* __restrict__ idx, int* __restrict__ count,
                           int B, int Npts, int Mtgt) {
  const int i = blockIdx.x * blockDim.x + threadIdx.x;
  if (i < B * Npts) {
    const int b = i / Npts;
    atomicAdd(&count[(size_t)b * Mtgt + idx[i]], 1);
  }
}

// sum over (b,n) of [1 - exp(-alpha*d) * frac/(count[nn]+eps)] in double.
__global__ void dacd_loss(const float* __restrict__ dist, const int* __restrict__ idx,
                          const int* __restrict__ count, double frac,
                          int B, int Npts, int Mtgt, double* __restrict__ sumAcc) {
  const int i = blockIdx.x * blockDim.x + threadIdx.x;
  double term = 0.0;
  if (i < B * Npts) {
    const int b = i / Npts;
    const double d = (double)dist[i];
    const double e = exp(-d * DACD_ALPHA);
    const double w = frac / ((double)count[(size_t)b * Mtgt + idx[i]] + DACD_EPS);
    term = 1.0 - e * w;
  }
#pragma unroll
  for (int off = 16; off >= 1; off >>= 1)
    term += __shfl_xor(term, off, 32);              // wave32 tree sum
  if ((threadIdx.x & 31) == 0)
    atomicAdd(sumAcc, term);                        // global_atomic_add_f64
}

__global__ void dacd_final(const double* __restrict__ sums, float* __restrict__ out,
                           int B, int N, int M) {
  if (blockIdx.x == 0 && threadIdx.x == 0) {
    const double loss = (sums[0] / (double)N + sums[1] / (double)M) / (2.0 * (double)B);
    out[0] = (float)loss;
  }
}

extern "C" void kernel_launch(void* const* d_in, const int* in_sizes, int n_in,
                              void* d_out, int out_size, void* d_ws, size_t ws_size,
                              hipStream_t stream) {
  const float* x  = (const float*)d_in[0];   // [B,3,N]
  const float* gt = (const float*)d_in[1];   // [B,3,M]
  const int B = 2, D = 3;
  const int N = in_sizes[0] / (B * D);       // 8192
  const int M = in_sizes[1] / (B * D);       // 8192
  float* out = (float*)d_out;

  // Workspace carve-out (all chunks are multiples of 8 bytes here).
  char* ws = (char*)d_ws;
  float* dist1  = (float*)ws;  ws += (size_t)B * N * sizeof(float);
  int*   idx1   = (int*)ws;    ws += (size_t)B * N * sizeof(int);
  float* dist2  = (float*)ws;  ws += (size_t)B * M * sizeof(float);
  int*   idx2   = (int*)ws;    ws += (size_t)B * M * sizeof(int);
  int*   count1 = (int*)ws;    ws += (size_t)B * M * sizeof(int);   // NN counts on gt side
  int*   count2 = (int*)ws;    ws += (size_t)B * N * sizeof(int);   // NN counts on x side
  double* sums  = (double*)ws;                                      // [2] accumulators

  const int initN = (B * N > B * M ? B * N : B * M);
  dacd_init<<<(initN + 255) / 256, 256, 0, stream>>>(count1, count2, sums, B * M, B * N);

  const int wavesPerBlock = 8;               // 256 threads = 8 wave32
  const int tiles1 = B * (N / 16);
  const int tiles2 = B * (M / 16);
  dacd_nn_wmma<<<(tiles1 + wavesPerBlock - 1) / wavesPerBlock, 32 * wavesPerBlock, 0, stream>>>(
      x, gt, dist1, idx1, B, N, M);
  dacd_nn_wmma<<<(tiles2 + wavesPerBlock - 1) / wavesPerBlock, 32 * wavesPerBlock, 0, stream>>>(
      gt, x, dist2, idx2, B, M, N);

  dacd_count<<<(B * N + 255) / 256, 256, 0, stream>>>(idx1, count1, B, N, M);
  dacd_count<<<(B * M + 255) / 256, 256, 0, stream>>>(idx2, count2, B, M, N);

  // NON_REG == False: frac_12 = N/M, frac_21 = M/N.
  const double frac12 = (double)N / (double)M;
  const double frac21 = (double)M / (double)N;
  dacd_loss<<<(B * N + 255) / 256, 256, 0, stream>>>(dist1, idx1, count1, frac21, B, N, M, &sums[0]);
  dacd_loss<<<(B * M + 255) / 256, 256, 0, stream>>>(dist2, idx2, count2, frac12, B, M, N, &sums[1]);

  dacd_final<<<1, 32, 0, stream>>>(sums, out, B, N, M);
}